// GNNModel_59038620451285
// MI455X (gfx1250) — compile-verified
//
#include <hip/hip_runtime.h>
#include <hip/hip_bf16.h>
#include <cstdint>

#define N_NODES  100000
#define N_EDGES  1600000
#define N_GRAPHS 1024
#define HID      128
#define LAYERS   4
#define LN_EPS   1e-5f
#define LDW      132   // LDS row stride (floats) for A tile: bank-conflict-free b64 frags
#define LDP      288   // LDS stride (floats) per K-pair row of W: half-wave offset = 128B mod 256B

typedef float v2f __attribute__((ext_vector_type(2)));
typedef float v8f __attribute__((ext_vector_type(8)));
typedef int   v4i __attribute__((vector_size(16)));

#if defined(__HIP_DEVICE_COMPILE__) && __has_builtin(__builtin_amdgcn_global_load_async_to_lds_b128)
#define HAVE_ASYNC_LDS 1
#define AS1P(p) ((__attribute__((address_space(1))) v4i*)(uintptr_t)(p))
#define AS3P(p) ((__attribute__((address_space(3))) v4i*)(uint32_t)(uintptr_t)(p))
#endif

__device__ __forceinline__ void wait_async_zero() {
#if defined(__HIP_DEVICE_COMPILE__)
#if __has_builtin(__builtin_amdgcn_s_wait_asynccnt)
    __builtin_amdgcn_s_wait_asynccnt(0);
#else
    asm volatile("s_wait_asynccnt 0x0" ::: "memory");
#endif
#endif
}

// ---------------------------------------------------------------------------
// zero / copy helpers (float4 granularity)
// ---------------------------------------------------------------------------
__global__ void zero_kernel(float* __restrict__ p, int n4) {
    int i = blockIdx.x * blockDim.x + threadIdx.x;
    if (i < n4) ((float4*)p)[i] = make_float4(0.f, 0.f, 0.f, 0.f);
}

__global__ void copy_kernel(float* __restrict__ dst, const float* __restrict__ src, int n4) {
    int i = blockIdx.x * blockDim.x + threadIdx.x;
    if (i < n4) ((float4*)dst)[i] = ((const float4*)src)[i];
}

// ---------------------------------------------------------------------------
// input projection: h[n,:] = x[n,0:7] @ inW + inb   (one block per node)
// ---------------------------------------------------------------------------
__global__ void proj_kernel(const float* __restrict__ x,
                            const float* __restrict__ inW,
                            const float* __restrict__ inb,
                            float* __restrict__ h) {
    int n = blockIdx.x, c = threadIdx.x;
    __shared__ float xr[7];
    if (c < 7) xr[c] = x[n * 7 + c];
    __syncthreads();
    float s = inb[c];
#pragma unroll
    for (int d = 0; d < 7; ++d) s += xr[d] * inW[d * HID + c];
    h[n * HID + c] = s;
}

// ---------------------------------------------------------------------------
// edge phase: msg = relu(h[src] + edge_attr@eW + eb); agg[dst] += msg
// one wave per edge, 4 channels per lane, float4 gather, f32 atomic scatter.
// (agg is pre-initialized to h, so after this kernel agg == g = sum + h.)
// ---------------------------------------------------------------------------
__global__ __launch_bounds__(256)
void edge_kernel(const int* __restrict__ ei,
                 const float* __restrict__ ea,
                 const float* __restrict__ h,
                 float* __restrict__ agg,
                 const float* __restrict__ eW,
                 const float* __restrict__ eb) {
    __shared__ float sW[3 * HID];
    __shared__ float sB[HID];
    int t = threadIdx.x;
    for (int i = t; i < 3 * HID; i += 256) sW[i] = eW[i];
    if (t < HID) sB[t] = eb[t];
    __syncthreads();

    int gid  = blockIdx.x * 256 + t;
    int e    = gid >> 5;
    int lane = t & 31;
    int c0   = lane * 4;
    if (e >= N_EDGES) return;

    int src = ei[e];
    int dst = ei[N_EDGES + e];
    float a0 = ea[e * 3 + 0], a1 = ea[e * 3 + 1], a2 = ea[e * 3 + 2];

    float4 hv = ((const float4*)h)[src * (HID / 4) + lane];
    float4 w0 = *(const float4*)&sW[0 * HID + c0];
    float4 w1 = *(const float4*)&sW[1 * HID + c0];
    float4 w2 = *(const float4*)&sW[2 * HID + c0];
    float4 bb = *(const float4*)&sB[c0];

    float m0 = fmaxf(hv.x + bb.x + a0 * w0.x + a1 * w1.x + a2 * w2.x, 0.f);
    float m1 = fmaxf(hv.y + bb.y + a0 * w0.y + a1 * w1.y + a2 * w2.y, 0.f);
    float m2 = fmaxf(hv.z + bb.z + a0 * w0.z + a1 * w1.z + a2 * w2.z, 0.f);
    float m3 = fmaxf(hv.w + bb.w + a0 * w0.w + a1 * w1.w + a2 * w2.w, 0.f);

    float* ap = &agg[dst * HID + c0];
    atomicAdd(ap + 0, m0);
    atomicAdd(ap + 1, m1);
    atomicAdd(ap + 2, m2);
    atomicAdd(ap + 3, m3);
}

// ---------------------------------------------------------------------------
// fused GEMM: out = relu(A @ W + bias), A:[nrows,128], W:[128,128]
// 256 threads = 8 waves; block covers 128 rows; wave computes a 16x128 stripe
// via V_WMMA_F32_16X16X4_F32 (32 K-chunks x 8 N-tiles).
//   - A tile staged global->LDS via GLOBAL_LOAD_ASYNC_TO_LDS_B128 (if avail)
//   - W staged K-pair-interleaved so each B frag is one aligned ds_load_b64
// ---------------------------------------------------------------------------
__global__ __launch_bounds__(256)
void gemm_relu_kernel(const float* __restrict__ A,
                      const float* __restrict__ W,
                      const float* __restrict__ bias,
                      float* __restrict__ out,
                      int nrows) {
    __shared__ float sW[64 * LDP];    // K-pair p: sW[p*LDP + col*2 + q] = W[2p+q][col]
    __shared__ float sA[HID * LDW];   // row-major A tile, padded stride
    __shared__ float sB[HID];

    const int t = threadIdx.x;
    const int blockRow0 = blockIdx.x * 128;

    // ---- stage W into K-pair interleaved layout (64 p-rows x 128 col-pairs)
    {
        // 8192 (p,col) entries, 32 per thread; consecutive threads -> consecutive cols
        for (int i = 0; i < 32; ++i) {
            int linear = i * 256 + t;           // coalesced across threads
            int p   = linear >> 7;              // 0..63
            int col = linear & 127;
            v2f w;
            w.x = W[(2 * p + 0) * HID + col];
            w.y = W[(2 * p + 1) * HID + col];
            *(v2f*)&sW[p * LDP + col * 2] = w;
        }
    }
    if (t < HID) sB[t] = bias[t];

    // ---- stage A tile (async global->LDS when supported)
    {
        int tr = t >> 1;                        // row within block tile: 0..127
        int ch = (t & 1) * 64;                  // column half
        int grow = blockRow0 + tr;
#pragma unroll
        for (int i = 0; i < 16; ++i) {
            int col = ch + i * 4;
            if (grow < nrows) {
#ifdef HAVE_ASYNC_LDS
                __builtin_amdgcn_global_load_async_to_lds_b128(
                    AS1P(A + (size_t)grow * HID + col),
                    AS3P(&sA[tr * LDW + col]), 0, 0);
#else
                float4 v = ((const float4*)A)[grow * (HID / 4) + (col >> 2)];
                *(float4*)&sA[tr * LDW + col] = v;
#endif
            } else {
                *(float4*)&sA[tr * LDW + col] = make_float4(0.f, 0.f, 0.f, 0.f);
            }
        }
    }
#ifdef HAVE_ASYNC_LDS
    wait_async_zero();
#endif
    __syncthreads();

    const int wv   = t >> 5;
    const int lane = t & 31;
    const int half = lane >> 4;   // K-half selector for A/B frags
    const int lm   = lane & 15;

    const float* ap    = &sA[(wv * 16 + lm) * LDW];
    const float* wbase = &sW[half * LDP + lm * 2];

    v8f acc[8];
#pragma unroll
    for (int i = 0; i < 8; ++i) acc[i] = (v8f){};

    for (int k = 0; k < 32; ++k) {            // K in chunks of 4
        v2f a = *(const v2f*)(ap + 4 * k + 2 * half);
        const float* wp = wbase + (2 * k) * LDP;   // pair row p = 2k + half
#pragma unroll
        for (int tt = 0; tt < 8; ++tt) {      // 8 N-tiles of 16 cols
            v2f b = *(const v2f*)(wp + tt * 32);
            acc[tt] = __builtin_amdgcn_wmma_f32_16x16x4_f32(
                false, a, false, b, (short)0, acc[tt], false, false);
        }
    }

#pragma unroll
    for (int tt = 0; tt < 8; ++tt) {
        int col = tt * 16 + lm;
        float bb = sB[col];
#pragma unroll
        for (int j = 0; j < 8; ++j) {
            int row = blockRow0 + wv * 16 + j + 8 * half;
            if (row < nrows)
                out[row * HID + col] = fmaxf(acc[tt][j] + bb, 0.f);
        }
    }
}

// ---------------------------------------------------------------------------
// LayerNorm: h = LN(h + u) * gamma + beta   (u already relu'd by GEMM2)
// ---------------------------------------------------------------------------
__global__ void ln_kernel(float* __restrict__ h,
                          const float* __restrict__ u,
                          const float* __restrict__ g,
                          const float* __restrict__ b) {
    int n = blockIdx.x, c = threadIdx.x;
    float v = h[n * HID + c] + u[n * HID + c];
    float s = v, sq = v * v;
#pragma unroll
    for (int m = 16; m >= 1; m >>= 1) {
        s  += __shfl_xor(s, m, 32);
        sq += __shfl_xor(sq, m, 32);
    }
    __shared__ float ss[4], ssq[4];
    int w = c >> 5;
    if ((c & 31) == 0) { ss[w] = s; ssq[w] = sq; }
    __syncthreads();
    s  = ss[0] + ss[1] + ss[2] + ss[3];
    sq = ssq[0] + ssq[1] + ssq[2] + ssq[3];
    float mu  = s / (float)HID;
    float var = sq / (float)HID - mu * mu;
    h[n * HID + c] = (v - mu) * rsqrtf(var + LN_EPS) * g[c] + b[c];
}

// ---------------------------------------------------------------------------
// global mean pool accumulation: sums[batch[n]] += h[n], cnt[batch[n]] += 1
// ---------------------------------------------------------------------------
__global__ __launch_bounds__(256)
void pool_kernel(const float* __restrict__ h,
                 const int* __restrict__ batch,
                 float* __restrict__ sums,
                 float* __restrict__ cnt) {
    int gid  = blockIdx.x * 256 + threadIdx.x;
    int n    = gid >> 5;
    int lane = threadIdx.x & 31;
    if (n >= N_NODES) return;
    int b = batch[n];
    float4 hv = ((const float4*)h)[n * (HID / 4) + lane];
    float* sp = &sums[b * HID + lane * 4];
    atomicAdd(sp + 0, hv.x);
    atomicAdd(sp + 1, hv.y);
    atomicAdd(sp + 2, hv.z);
    atomicAdd(sp + 3, hv.w);
    if (lane == 0) atomicAdd(&cnt[b], 1.0f);
}

// ---------------------------------------------------------------------------
// head MLP per graph: 128 -> 128 -> 64 -> 1
// ---------------------------------------------------------------------------
__global__ void head_kernel(const float* __restrict__ sums,
                            const float* __restrict__ cnt,
                            const float* __restrict__ fcW1, const float* __restrict__ fcb1,
                            const float* __restrict__ fcW2, const float* __restrict__ fcb2,
                            const float* __restrict__ fcW3, const float* __restrict__ fcb3,
                            float* __restrict__ out) {
    __shared__ float p[HID], o1[HID], o2[HID / 2];
    int g = blockIdx.x, c = threadIdx.x;
    float cn = fmaxf(cnt[g], 1.0f);
    p[c] = sums[g * HID + c] / cn;
    __syncthreads();
    float s = fcb1[c];
    for (int k = 0; k < HID; ++k) s += p[k] * fcW1[k * HID + c];
    o1[c] = fmaxf(s, 0.f);
    __syncthreads();
    if (c < HID / 2) {
        float s2 = fcb2[c];
        for (int k = 0; k < HID; ++k) s2 += o1[k] * fcW2[k * (HID / 2) + c];
        o2[c] = fmaxf(s2, 0.f);
    }
    __syncthreads();
    if (c == 0) {
        float s3 = fcb3[0];
        for (int k = 0; k < HID / 2; ++k) s3 += o2[k] * fcW3[k];
        out[g] = s3;
    }
}

// ---------------------------------------------------------------------------
extern "C" void kernel_launch(void* const* d_in, const int* in_sizes, int n_in,
                              void* d_out, int out_size, void* d_ws, size_t ws_size,
                              hipStream_t stream) {
    const float* x          = (const float*)d_in[0];
    const int*   edge_index = (const int*)  d_in[1];
    const float* edge_attr  = (const float*)d_in[2];
    const int*   batch      = (const int*)  d_in[3];
    const float* inW        = (const float*)d_in[4];
    const float* inb        = (const float*)d_in[5];
    const float* edgeW      = (const float*)d_in[6];
    const float* edgeb      = (const float*)d_in[7];
    const float* w1         = (const float*)d_in[8];
    const float* b1         = (const float*)d_in[9];
    const float* w2         = (const float*)d_in[10];
    const float* b2         = (const float*)d_in[11];
    const float* gamma      = (const float*)d_in[12];
    const float* beta       = (const float*)d_in[13];
    const float* fcW1       = (const float*)d_in[14];
    const float* fcb1       = (const float*)d_in[15];
    const float* fcW2       = (const float*)d_in[16];
    const float* fcb2       = (const float*)d_in[17];
    const float* fcW3       = (const float*)d_in[18];
    const float* fcb3       = (const float*)d_in[19];
    float* out = (float*)d_out;

    const size_t NH = (size_t)N_NODES * HID;           // 12.8M floats
    float* h    = (float*)d_ws;
    float* agg  = h + NH;
    float* t1   = agg + NH;
    float* sums = t1;                                   // reuse t1 after layers
    float* cnt  = sums + (size_t)N_GRAPHS * HID;

    const int gemmBlocks = (N_NODES + 127) / 128;       // 782
    const int edgeBlocks = (N_EDGES * 32 + 255) / 256;  // 200000
    const int nh4Blocks  = ((int)(NH / 4) + 255) / 256;

    proj_kernel<<<N_NODES, HID, 0, stream>>>(x, inW, inb, h);

    for (int l = 0; l < LAYERS; ++l) {
        // agg := h  (so the edge scatter accumulates directly into g = sum + h)
        copy_kernel<<<nh4Blocks, 256, 0, stream>>>(agg, h, (int)(NH / 4));
        edge_kernel<<<edgeBlocks, 256, 0, stream>>>(
            edge_index, edge_attr, h, agg,
            edgeW + (size_t)l * 3 * HID, edgeb + (size_t)l * HID);
        // t1 = relu(g @ w1 + b1)
        gemm_relu_kernel<<<gemmBlocks, 256, 0, stream>>>(
            agg, w1 + (size_t)l * HID * HID, b1 + (size_t)l * HID, t1, N_NODES);
        // agg = relu(t1 @ w2 + b2)  == relu(g_mlp); agg buffer is dead here
        gemm_relu_kernel<<<gemmBlocks, 256, 0, stream>>>(
            t1, w2 + (size_t)l * HID * HID, b2 + (size_t)l * HID, agg, N_NODES);
        // h = LN(h + relu(g_mlp))
        ln_kernel<<<N_NODES, HID, 0, stream>>>(h, agg, gamma + (size_t)l * HID,
                                               beta + (size_t)l * HID);
    }

    // pooling: zero sums+cnt (contiguous), accumulate, head
    int poolN4 = (N_GRAPHS * HID + N_GRAPHS) / 4;       // 33024 float4s
    zero_kernel<<<(poolN4 + 255) / 256, 256, 0, stream>>>(sums, poolN4);
    pool_kernel<<<(N_NODES * 32 + 255) / 256, 256, 0, stream>>>(h, batch, sums, cnt);
    head_kernel<<<N_GRAPHS, HID, 0, stream>>>(sums, cnt, fcW1, fcb1, fcW2, fcb2,
                                              fcW3, fcb3, out);
}